// APGCN_8735963480652
// MI455X (gfx1250) — compile-verified
//
#include <hip/hip_runtime.h>

#define F_IN  512
#define F_HID 256
#define NC    64
#define NITER 10

typedef float v2f __attribute__((ext_vector_type(2)));
typedef float v8f __attribute__((ext_vector_type(8)));

// ---------------------------------------------------------------------------
// Fused MLP: local = relu(x@W1 + b1) @ W2 + b2
// Block = (32,4) = 4 waves, one 16-row node tile per block.
// Phase 1: each wave computes 4 hidden 16x16 tiles (K=512) -> LDS (16x256).
// Phase 2: each wave computes one output 16x16 tile (K=256 from LDS).
// ---------------------------------------------------------------------------
__global__ __launch_bounds__(128)
void apgcn_mlp_fused(const float* __restrict__ x,
                     const float* __restrict__ W1, const float* __restrict__ b1,
                     const float* __restrict__ W2, const float* __restrict__ b2,
                     float* __restrict__ local, int nnodes)
{
    __shared__ float hid[16][F_HID];   // 16 KB of the 320 KB WGP LDS

    const int lane = threadIdx.x;       // 0..31
    const int wave = threadIdx.y;       // 0..3
    const int m0   = blockIdx.x * 16;
    const int row  = lane & 15;
    const int half = lane >> 4;         // 0 or 1

    int gr = m0 + row;
    if (gr >= nnodes) gr = nnodes - 1;  // uniform-safe clamp (N % 16 == 0 anyway)
    const float* xrow = x + (size_t)gr * F_IN;

    // ---- Phase 1: hidden tiles, n-cols [wave*64, wave*64+64) ----
    for (int t = 0; t < 4; ++t) {
        const int n0 = (wave * 4 + t) * 16;
        v8f c = {};
        for (int k0 = 0; k0 < F_IN; k0 += 4) {
            const int ka = k0 + 2 * half;
            v2f a, b;
            a.x = xrow[ka];
            a.y = xrow[ka + 1];
            b.x = W1[(size_t)(ka    ) * F_HID + n0 + row];
            b.y = W1[(size_t)(ka + 1) * F_HID + n0 + row];
            c = __builtin_amdgcn_wmma_f32_16x16x4_f32(false, a, false, b,
                                                      (short)0, c, false, false);
        }
        const float bias = b1[n0 + row];
        #pragma unroll
        for (int r = 0; r < 8; ++r) {
            const int mm = r + 8 * half;
            float v = c[r] + bias;
            hid[mm][n0 + row] = v > 0.0f ? v : 0.0f;
        }
    }
    __syncthreads();

    // ---- Phase 2: output tile, n-cols [wave*16, wave*16+16), K=256 ----
    {
        const int n0 = wave * 16;
        v8f c = {};
        for (int k0 = 0; k0 < F_HID; k0 += 4) {
            const int ka = k0 + 2 * half;
            v2f a, b;
            a.x = hid[row][ka];
            a.y = hid[row][ka + 1];
            b.x = W2[(size_t)(ka    ) * NC + n0 + row];
            b.y = W2[(size_t)(ka + 1) * NC + n0 + row];
            c = __builtin_amdgcn_wmma_f32_16x16x4_f32(false, a, false, b,
                                                      (short)0, c, false, false);
        }
        const float bias = b2[n0 + row];
        #pragma unroll
        for (int r = 0; r < 8; ++r) {
            const int mm = r + 8 * half;
            if (m0 + mm < nnodes)
                local[(size_t)(m0 + mm) * NC + n0 + row] = c[r] + bias;
        }
    }
}

// ---------------------------------------------------------------------------
// Init: xacc=0 over n*64; per-node state; deg=1 (self loop pre-counted)
// ---------------------------------------------------------------------------
__global__ void apgcn_init(float* __restrict__ xacc, float* __restrict__ steps,
                           float* __restrict__ sumh, float* __restrict__ cont,
                           float* __restrict__ deg, int n, int tot)
{
    int i = blockIdx.x * 256 + threadIdx.x;
    if (i < tot) xacc[i] = 0.0f;
    if (i < n) {
        steps[i] = 1.0f;
        sumh[i]  = 0.0f;
        cont[i]  = 1.0f;
        deg[i]   = 1.0f;   // self-loop contribution to degree
    }
}

__global__ void apgcn_deg(const long long* __restrict__ ei, float* __restrict__ deg, int e)
{
    int i = blockIdx.x * 256 + threadIdx.x;
    if (i >= e) return;
    int d = (int)ei[(size_t)e + i];     // dst row
    atomicAdd(&deg[d], 1.0f);
}

__global__ void apgcn_dis(float* __restrict__ deg, int n)
{
    int i = blockIdx.x * 256 + threadIdx.x;
    if (i < n) deg[i] = rsqrtf(deg[i]);   // deg >= 1 always (self loop)
}

__global__ void apgcn_norm(const long long* __restrict__ ei, const float* __restrict__ dis,
                           float* __restrict__ normw, int e)
{
    int i = blockIdx.x * 256 + threadIdx.x;
    if (i >= e) return;
    int s = (int)ei[i];
    int d = (int)ei[(size_t)e + i];
    normw[i] = dis[s] * dis[d];
}

// ---------------------------------------------------------------------------
// Propagation: nxt = selfnorm*cur (init)  then  nxt[dst] += norm*cur[src]
// ---------------------------------------------------------------------------
__global__ void apgcn_prop_self(const float* __restrict__ cur, const float* __restrict__ dis,
                                float* __restrict__ nxt, int n)
{
    int idx = blockIdx.x * 256 + threadIdx.x;
    if (idx >= n * 16) return;
    int node = idx >> 4;
    int c4   = (idx & 15) * 4;
    float w = dis[node]; w = w * w;                 // dis[i]^2 = 1/deg
    const float4 v = *(const float4*)(cur + (size_t)node * NC + c4);
    float4 r; r.x = w * v.x; r.y = w * v.y; r.z = w * v.z; r.w = w * v.w;
    *(float4*)(nxt + (size_t)node * NC + c4) = r;
}

__global__ void apgcn_scatter(const long long* __restrict__ ei, const float* __restrict__ normw,
                              const float* __restrict__ cur, float* __restrict__ nxt, int e)
{
    int idx = blockIdx.x * 256 + threadIdx.x;
    if (idx >= e * 16) return;
    int ed = idx >> 4;
    int c4 = (idx & 15) * 4;
    int s = (int)ei[ed];
    int d = (int)ei[(size_t)e + ed];
    float w = normw[ed];
    const float4 v = *(const float4*)(cur + (size_t)s * NC + c4);
    float* o = nxt + (size_t)d * NC + c4;
    atomicAdd(o + 0, w * v.x);
    atomicAdd(o + 1, w * v.y);
    atomicAdd(o + 2, w * v.z);
    atomicAdd(o + 3, w * v.w);
}

// ---------------------------------------------------------------------------
// Halting + adaptive accumulation. One wave32 per node, 2 classes per lane.
// ---------------------------------------------------------------------------
__global__ __launch_bounds__(256)
void apgcn_halt(const float* __restrict__ nxt, const float* __restrict__ oldp,
                const float* __restrict__ Wh, const float* __restrict__ bh,
                float* __restrict__ steps, float* __restrict__ sumh,
                float* __restrict__ cont, float* __restrict__ xacc,
                int n, int is_last)
{
    const int lane = threadIdx.x & 31;
    const int node = blockIdx.x * 8 + (threadIdx.x >> 5);
    if (node >= n) return;

    const float2 v  = *(const float2*)(nxt  + (size_t)node * NC + lane * 2);
    const float2 vo = *(const float2*)(oldp + (size_t)node * NC + lane * 2);

    float dot = v.x * Wh[lane * 2] + v.y * Wh[lane * 2 + 1];
    #pragma unroll
    for (int off = 16; off; off >>= 1) dot += __shfl_xor(dot, off, 32);

    const float h  = 1.0f / (1.0f + __expf(-(dot + bh[0])));
    float sh = sumh[node];
    float st = steps[node];
    const float co = cont[node];

    const bool  pm = ((sh + h) < 0.99f) && (co != 0.0f);
    const float pf = pm ? 1.0f : 0.0f;
    st += pf * (is_last ? 0.0f : 1.0f);
    sh += pf * h;
    const bool cond = pm && (st < (float)NITER);
    const float p = cond ? sh : (1.0f - sh);

    float* xa = xacc + (size_t)node * NC + lane * 2;
    xa[0] += (p * v.x + (1.0f - p) * vo.x) * co;
    xa[1] += (p * v.y + (1.0f - p) * vo.y) * co;

    if (lane == 0) {
        steps[node] = st;
        sumh[node]  = sh;
        cont[node]  = pf;   // cont & prob_mask == prob_mask (pm already includes cont)
    }
}

// ---------------------------------------------------------------------------
// Finalize: out = log_softmax(xacc/steps); plus steps and 1-sum_h tails.
// ---------------------------------------------------------------------------
__global__ __launch_bounds__(256)
void apgcn_final(const float* __restrict__ xacc, const float* __restrict__ steps,
                 const float* __restrict__ sumh, float* __restrict__ out, int n)
{
    const int lane = threadIdx.x & 31;
    const int node = blockIdx.x * 8 + (threadIdx.x >> 5);
    if (node >= n) return;

    const float st = steps[node];
    float2 v = *(const float2*)(xacc + (size_t)node * NC + lane * 2);
    v.x /= st; v.y /= st;

    float m = fmaxf(v.x, v.y);
    #pragma unroll
    for (int off = 16; off; off >>= 1) m = fmaxf(m, __shfl_xor(m, off, 32));
    float s = __expf(v.x - m) + __expf(v.y - m);
    #pragma unroll
    for (int off = 16; off; off >>= 1) s += __shfl_xor(s, off, 32);
    const float lse = m + __logf(s);

    float* o = out + (size_t)node * NC + lane * 2;
    o[0] = v.x - lse;
    o[1] = v.y - lse;
    if (lane == 0) {
        out[(size_t)n * NC + node]     = st;
        out[(size_t)n * NC + n + node] = 1.0f - sumh[node];
    }
}

// ---------------------------------------------------------------------------
extern "C" void kernel_launch(void* const* d_in, const int* in_sizes, int n_in,
                              void* d_out, int out_size, void* d_ws, size_t ws_size,
                              hipStream_t stream)
{
    const float*     x  = (const float*)d_in[0];
    const long long* ei = (const long long*)d_in[1];   // int64 edge_index (2,E)
    const float*     W1 = (const float*)d_in[2];
    const float*     b1 = (const float*)d_in[3];
    const float*     W2 = (const float*)d_in[4];
    const float*     b2 = (const float*)d_in[5];
    const float*     Wh = (const float*)d_in[6];
    const float*     bh = (const float*)d_in[7];
    float* out = (float*)d_out;

    const int n = in_sizes[0] / F_IN;   // 100000
    const int e = in_sizes[1] / 2;      // 3200000

    // Workspace carving (floats): ~91 MB total
    float* ws = (float*)d_ws;
    size_t off = 0;
    float* local = ws + off; off += (size_t)n * NC;   // prop ping
    float* propB = ws + off; off += (size_t)n * NC;   // prop pong
    float* xacc  = ws + off; off += (size_t)n * NC;
    float* dis   = ws + off; off += (size_t)n;        // deg -> rsqrt(deg)
    float* normw = ws + off; off += (size_t)e;
    float* steps = ws + off; off += (size_t)n;
    float* sumh  = ws + off; off += (size_t)n;
    float* cont  = ws + off; off += (size_t)n;

    const int tot = n * NC;
    apgcn_init<<<(tot + 255) / 256, 256, 0, stream>>>(xacc, steps, sumh, cont, dis, n, tot);
    apgcn_deg <<<(e + 255) / 256, 256, 0, stream>>>(ei, dis, e);
    apgcn_dis <<<(n + 255) / 256, 256, 0, stream>>>(dis, n);
    apgcn_norm<<<(e + 255) / 256, 256, 0, stream>>>(ei, dis, normw, e);

    apgcn_mlp_fused<<<dim3((n + 15) / 16), dim3(32, 4), 0, stream>>>(
        x, W1, b1, W2, b2, local, n);

    float* cur = local;
    float* nxt = propB;
    for (int i = 0; i < NITER; ++i) {
        apgcn_prop_self<<<(n * 16 + 255) / 256, 256, 0, stream>>>(cur, dis, nxt, n);
        apgcn_scatter  <<<(e * 16 + 255) / 256, 256, 0, stream>>>(ei, normw, cur, nxt, e);
        apgcn_halt     <<<(n + 7) / 8,       256, 0, stream>>>(nxt, cur, Wh, bh,
                                                               steps, sumh, cont, xacc,
                                                               n, i == NITER - 1 ? 1 : 0);
        float* t = cur; cur = nxt; nxt = t;
    }

    apgcn_final<<<(n + 7) / 8, 256, 0, stream>>>(xacc, steps, sumh, out, n);
}